// HTop2Gate_56358560858467
// MI455X (gfx1250) — compile-verified
//
#include <hip/hip_runtime.h>
#include <math.h>

typedef float v2f __attribute__((ext_vector_type(2)));
typedef float v4f __attribute__((ext_vector_type(4)));
typedef float v8f __attribute__((ext_vector_type(8)));

// Problem constants (fixed by reference setup_inputs)
constexpr int S   = 4096;   // tokens
constexpr int HD  = 1024;   // hidden
constexpr int E   = 64;     // experts
constexpr int CAP = 512;    // capacity = 2*ceil(4096/16)
constexpr size_t NSEC = (size_t)S * E * CAP;      // 134217728 elems per output tensor

// ---------------------------------------------------------------------------
// 0) Zero-fill d_out (1 + 2*NSEC floats). Store-bound: ~1.07GB at 23.3TB/s.
// ---------------------------------------------------------------------------
__global__ __launch_bounds__(256) void fill_zero(v4f* __restrict__ p, size_t n4) {
    size_t i = (size_t)blockIdx.x * blockDim.x + threadIdx.x;
    size_t stride = (size_t)gridDim.x * blockDim.x;
    v4f z = {0.f, 0.f, 0.f, 0.f};
    for (; i < n4; i += stride) p[i] = z;
    if (blockIdx.x == 0 && threadIdx.x == 0) {
        ((float*)p)[4 * n4] = 0.0f;   // odd tail element (total = 1 + 2*NSEC)
    }
}

// ---------------------------------------------------------------------------
// 1) logits[S,E] = token_input[S,HD] @ W[E,HD]^T via V_WMMA_F32_16X16X4_F32.
//    Block: 128 threads = 4 waves. Block tile: 16 tokens x 64 experts.
//    K staged through LDS in 128-wide chunks; padded rows (132) avoid bank
//    conflicts for the strided v2f operand reads.
// ---------------------------------------------------------------------------
__global__ __launch_bounds__(128) void gemm_logits(const float* __restrict__ A,
                                                   const float* __restrict__ W,
                                                   float* __restrict__ logits) {
    __shared__ float As[16][132];
    __shared__ float Bs[64][132];

    const int t    = threadIdx.x;
    const int wave = t >> 5;
    const int lane = t & 31;
    const int tok0 = blockIdx.x * 16;

    const int mrow  = lane & 15;          // A row / B expert-within-tile (lanes 0-15 vs 16-31)
    const int khalf = (lane >> 4) * 2;    // lanes 16-31 handle K+2,K+3
    const int eoff  = wave * 16 + mrow;   // expert row in Bs

    v8f acc = {};

    for (int k0 = 0; k0 < HD; k0 += 128) {
        // Stage A chunk: 16 rows x 128 cols, coalesced float4 loads
        #pragma unroll
        for (int j = 0; j < 4; j++) {
            int f = t + j * 128;                 // float4 index among 512
            int r = f >> 5;
            int c = (f & 31) << 2;
            v4f v = *(const v4f*)(A + (size_t)(tok0 + r) * HD + k0 + c);
            *(v4f*)&As[r][c] = v;
        }
        // Stage B (weights) chunk: 64 rows x 128 cols
        #pragma unroll
        for (int j = 0; j < 16; j++) {
            int f = t + j * 128;                 // float4 index among 2048
            int r = f >> 5;
            int c = (f & 31) << 2;
            v4f v = *(const v4f*)(W + (size_t)r * HD + k0 + c);
            *(v4f*)&Bs[r][c] = v;
        }
        __syncthreads();

        #pragma unroll
        for (int kk = 0; kk < 128; kk += 4) {
            v2f a = *(const v2f*)&As[mrow][kk + khalf];
            v2f b = *(const v2f*)&Bs[eoff][kk + khalf];
            acc = __builtin_amdgcn_wmma_f32_16x16x4_f32(
                /*neg_a=*/false, a, /*neg_b=*/false, b,
                /*c_mod=*/(short)0, acc, /*reuse_a=*/false, /*reuse_b=*/false);
        }
        __syncthreads();
    }

    // C layout: VGPR v -> M = v + 8*(lane/16), N = lane%16
    #pragma unroll
    for (int v = 0; v < 8; v++) {
        int m = v + ((lane >> 4) << 3);
        int n = lane & 15;
        logits[(size_t)(tok0 + m) * E + wave * 16 + n] = acc[v];
    }
}

// ---------------------------------------------------------------------------
// 2) Per-token gating: softmax(E=64), top-1, JAX threefry gumbel, masked top-2.
//    One wave per token; lane handles experts (lane, lane+32).
// ---------------------------------------------------------------------------
__device__ inline unsigned rotl32(unsigned x, int r) { return (x << r) | (x >> (32 - r)); }

__device__ inline void threefry2x32(unsigned k0, unsigned k1, unsigned x0, unsigned x1,
                                    unsigned& o0, unsigned& o1) {
    const unsigned ks2 = k0 ^ k1 ^ 0x1BD11BDAu;
    x0 += k0; x1 += k1;
    #pragma unroll
    for (int i = 0; i < 4; i++) { const int r[4]={13,15,26,6};  x0 += x1; x1 = rotl32(x1, r[i]); x1 ^= x0; }
    x0 += k1;  x1 += ks2 + 1u;
    #pragma unroll
    for (int i = 0; i < 4; i++) { const int r[4]={17,29,16,24}; x0 += x1; x1 = rotl32(x1, r[i]); x1 ^= x0; }
    x0 += ks2; x1 += k0 + 2u;
    #pragma unroll
    for (int i = 0; i < 4; i++) { const int r[4]={13,15,26,6};  x0 += x1; x1 = rotl32(x1, r[i]); x1 ^= x0; }
    x0 += k0;  x1 += k1 + 3u;
    #pragma unroll
    for (int i = 0; i < 4; i++) { const int r[4]={17,29,16,24}; x0 += x1; x1 = rotl32(x1, r[i]); x1 ^= x0; }
    x0 += k1;  x1 += ks2 + 4u;
    #pragma unroll
    for (int i = 0; i < 4; i++) { const int r[4]={13,15,26,6};  x0 += x1; x1 = rotl32(x1, r[i]); x1 ^= x0; }
    x0 += ks2; x1 += k0 + 5u;
    o0 = x0; o1 = x1;
}

__device__ inline float gumbel_at(unsigned i) {
    const unsigned H = (unsigned)(S * E) / 2u;   // 131072
    unsigned x0, x1, o0, o1;
    if (i < H) { x0 = i;     x1 = i + H; }
    else       { x0 = i - H; x1 = i;     }
    threefry2x32(0u, 42u, x0, x1, o0, o1);       // jax.random.key(42) -> (0, 42)
    unsigned bits = (i < H) ? o0 : o1;
    float f = __uint_as_float((bits >> 9) | 0x3f800000u) - 1.0f;   // [0,1)
    const float tiny = 1.17549435e-38f;
    float u = fmaxf(tiny, f * (1.0f - tiny) + tiny);
    return -logf(-logf(u));
}

__global__ __launch_bounds__(256) void gate_kernel(const float* __restrict__ logits,
                                                   float* __restrict__ gates,
                                                   int* __restrict__ idx1, int* __restrict__ idx2,
                                                   float* __restrict__ g1v, float* __restrict__ g2v) {
    const int wave = threadIdx.x >> 5;
    const int lane = threadIdx.x & 31;
    const int s = blockIdx.x * 8 + wave;

    const float* lrow = logits + (size_t)s * E;
    float a = lrow[lane];
    float b = lrow[lane + 32];

    // softmax
    float m = fmaxf(a, b);
    #pragma unroll
    for (int off = 16; off; off >>= 1) m = fmaxf(m, __shfl_xor(m, off, 32));
    float ea = expf(a - m), eb = expf(b - m);
    float sum = ea + eb;
    #pragma unroll
    for (int off = 16; off; off >>= 1) sum += __shfl_xor(sum, off, 32);
    float ga = ea / sum, gb = eb / sum;
    gates[(size_t)s * E + lane]      = ga;
    gates[(size_t)s * E + lane + 32] = gb;

    // top-1 argmax (first occurrence -> smaller index on tie)
    float v; int id;
    if (gb > ga) { v = gb; id = lane + 32; } else { v = ga; id = lane; }
    #pragma unroll
    for (int off = 16; off; off >>= 1) {
        float ov = __shfl_xor(v, off, 32);
        int   oi = __shfl_xor(id, off, 32);
        if (ov > v || (ov == v && oi < id)) { v = ov; id = oi; }
    }
    const int i1 = id;
    const float g1 = v;

    // gumbel-noised, top-1-masked argmax
    unsigned base = (unsigned)s * (unsigned)E;
    float na = ga + gumbel_at(base + lane);
    float nb = gb + gumbel_at(base + lane + 32);
    if (lane == i1)       na = -INFINITY;
    if (lane + 32 == i1)  nb = -INFINITY;
    if (nb > na) { v = nb; id = lane + 32; } else { v = na; id = lane; }
    #pragma unroll
    for (int off = 16; off; off >>= 1) {
        float ov = __shfl_xor(v, off, 32);
        int   oi = __shfl_xor(id, off, 32);
        if (ov > v || (ov == v && oi < id)) { v = ov; id = oi; }
    }
    const int i2 = id;
    float g2 = (i2 < 32) ? __shfl(ga, i2, 32) : __shfl(gb, i2 - 32, 32);

    if (lane == 0) {
        idx1[s] = i1; idx2[s] = i2;
        g1v[s] = g1;  g2v[s] = g2;
    }
}

// ---------------------------------------------------------------------------
// 3) Per-expert ordered scan: cumsum-based slot locations (deterministic,
//    ballot+LDS prefix), counts, and me*ce partial for l_aux.
// ---------------------------------------------------------------------------
__global__ __launch_bounds__(256) void scan_kernel(const int* __restrict__ idx1,
                                                   const int* __restrict__ idx2,
                                                   const float* __restrict__ gates,
                                                   int* __restrict__ loc1, int* __restrict__ loc2,
                                                   float* __restrict__ me_ce) {
    const int e = blockIdx.x;
    const int t = threadIdx.x;
    const int lane = t & 31, w = t >> 5;
    __shared__ int wsum[8];
    __shared__ float red[256];

    // pass 1: locations for top-1 assignments (token order)
    int running = 0;
    for (int c = 0; c < S / 256; c++) {
        int s = c * 256 + t;
        bool mflag = (idx1[s] == e);
        unsigned mask = (unsigned)__ballot(mflag);
        int pre = __popc(mask & ((1u << lane) - 1u));
        int wt  = __popc(mask);
        if (lane == 0) wsum[w] = wt;
        __syncthreads();
        int wpre = 0, tot = 0;
        #pragma unroll
        for (int i = 0; i < 8; i++) { if (i < w) wpre += wsum[i]; tot += wsum[i]; }
        if (mflag) loc1[s] = running + wpre + pre;
        running += tot;
        __syncthreads();
    }
    const int count1 = running;

    // pass 2: locations for top-2 assignments, offset by count1
    running = 0;
    for (int c = 0; c < S / 256; c++) {
        int s = c * 256 + t;
        bool mflag = (idx2[s] == e);
        unsigned mask = (unsigned)__ballot(mflag);
        int pre = __popc(mask & ((1u << lane) - 1u));
        int wt  = __popc(mask);
        if (lane == 0) wsum[w] = wt;
        __syncthreads();
        int wpre = 0, tot = 0;
        #pragma unroll
        for (int i = 0; i < 8; i++) { if (i < w) wpre += wsum[i]; tot += wsum[i]; }
        if (mflag) loc2[s] = count1 + running + wpre + pre;
        running += tot;
        __syncthreads();
    }

    // me[e] = mean_s gates[s,e]; ce[e] = count1/S  (deterministic tree reduce)
    float p = 0.f;
    for (int c = 0; c < S / 256; c++) p += gates[(size_t)(c * 256 + t) * E + e];
    red[t] = p;
    __syncthreads();
    for (int off = 128; off; off >>= 1) {
        if (t < off) red[t] += red[t + off];
        __syncthreads();
    }
    if (t == 0) {
        float me = red[0] / (float)S;
        float ce = (float)count1 / (float)S;
        me_ce[e] = me * ce;
    }
}

// ---------------------------------------------------------------------------
// 4a) l_aux = sum(me*ce) / num_2nd * (E*E) = sum * 256
// ---------------------------------------------------------------------------
__global__ __launch_bounds__(64) void laux_kernel(const float* __restrict__ me_ce,
                                                  float* __restrict__ out) {
    __shared__ float red[64];
    int t = threadIdx.x;
    red[t] = me_ce[t];
    __syncthreads();
    for (int off = 32; off; off >>= 1) {
        if (t < off) red[t] += red[t + off];
        __syncthreads();
    }
    if (t == 0) out[0] = red[0] * ((float)(E * E) / 16.0f);   // *256
}

// ---------------------------------------------------------------------------
// 4b) Capacity drop + renormalize + scatter the <=2 nonzeros per token.
// ---------------------------------------------------------------------------
__global__ __launch_bounds__(256) void scatter_kernel(const int* __restrict__ idx1,
                                                      const int* __restrict__ idx2,
                                                      const int* __restrict__ loc1,
                                                      const int* __restrict__ loc2,
                                                      const float* __restrict__ g1v,
                                                      const float* __restrict__ g2v,
                                                      float* __restrict__ out) {
    const int s = blockIdx.x * 256 + threadIdx.x;
    const int i1 = idx1[s], i2 = idx2[s];
    const int l1 = loc1[s], l2 = loc2[s];
    const bool k1 = (l1 < CAP);
    const bool k2 = (l2 < CAP);
    float g1s = k1 ? g1v[s] : 0.0f;
    float g2s = k2 ? g2v[s] : 0.0f;
    const float eps = 1.1920929e-7f;            // finfo(f32).eps
    float denom = fmaxf(g1s + g2s, eps);
    float g1n = g1s / denom;
    float g2n = g2s / denom;

    float* cw = out + 1;
    float* dm = cw + NSEC;
    if (k1 && g1n > 0.0f) {
        size_t o = ((size_t)s * E + i1) * CAP + l1;
        cw[o] = g1n; dm[o] = 1.0f;
    }
    if (k2 && g2n > 0.0f) {
        size_t o = ((size_t)s * E + i2) * CAP + l2;
        cw[o] = g2n; dm[o] = 1.0f;
    }
}

// ---------------------------------------------------------------------------
extern "C" void kernel_launch(void* const* d_in, const int* in_sizes, int n_in,
                              void* d_out, int out_size, void* d_ws, size_t ws_size,
                              hipStream_t stream) {
    const float* tok = (const float*)d_in[0];   // [4096,1024]
    const float* wg  = (const float*)d_in[1];   // [64,1024]

    char* ws = (char*)d_ws;
    float* logits = (float*)ws;                         // 1 MiB
    float* gates  = (float*)(ws + (1u << 20));          // 1 MiB
    int*   idx1   = (int*)(ws + (2u << 20));
    int*   idx2   = idx1 + S;
    float* g1v    = (float*)(idx2 + S);
    float* g2v    = g1v + S;
    int*   loc1   = (int*)(g2v + S);
    int*   loc2   = loc1 + S;
    float* me_ce  = (float*)(loc2 + S);

    float* out = (float*)d_out;

    const size_t n4 = (2 * NSEC) / 4;                   // 67108864 float4s from base
    fill_zero<<<4096, 256, 0, stream>>>((v4f*)out, n4);
    gemm_logits<<<S / 16, 128, 0, stream>>>(tok, wg, logits);
    gate_kernel<<<S / 8, 256, 0, stream>>>(logits, gates, idx1, idx2, g1v, g2v);
    scan_kernel<<<E, 256, 0, stream>>>(idx1, idx2, gates, loc1, loc2, me_ce);
    laux_kernel<<<1, 64, 0, stream>>>(me_ce, out);
    scatter_kernel<<<S / 256, 256, 0, stream>>>(idx1, idx2, loc1, loc2, g1v, g2v, out);
}